// VectorQuantizer_764504178920
// MI455X (gfx1250) — compile-verified
//
#include <hip/hip_runtime.h>
#include <hip/hip_bf16.h>

// Problem constants (from reference): B=2, C=32, T=8, H=32, W=32, V=8192
constexpr int DIM   = 32;      // embedding dim C
constexpr int VOCAB = 8192;    // codebook size
constexpr int NQ    = 16384;   // B*T*H*W query vectors
constexpr int NELEM = 524288;  // B*C*T*H*W output elements
constexpr int THW   = 8192;    // T*H*W
constexpr int NCHUNK = VOCAB / 16;   // 512 column chunks of 16 codes

typedef __attribute__((ext_vector_type(2))) float v2f;
typedef __attribute__((ext_vector_type(8))) float v8f;

// ---------------------------------------------------------------------------
// CDNA5 async global->LDS helpers.
// On CDNA5 the low 32 bits of a flat pointer into the LDS aperture are the raw
// LDS byte address (ISA 10.2).
// ---------------------------------------------------------------------------
__device__ __forceinline__ unsigned lds_addr_of(const void* p) {
    return (unsigned)(unsigned long long)p;
}

// GVS mode: mem = SGPR_base + VGPR_offset + IOFFSET ; LDS dest byte addr in VDST.
// Tracked with ASYNCcnt (not LOADcnt) -> overlaps with the WMMA chain.
__device__ __forceinline__ void async_stage_b64(unsigned lds_dst,
                                                const float* sbase,
                                                unsigned voff_bytes) {
    asm volatile("global_load_async_to_lds_b64 %0, %1, %2"
                 :
                 : "v"(lds_dst), "v"(voff_bytes), "s"(sbase)
                 : "memory");
}

__device__ __forceinline__ void wait_async0() {
    asm volatile("s_wait_asynccnt 0x0" ::: "memory");
}

// ---------------------------------------------------------------------------
// Kernel 1: codebook squared norms + zero the "used" flags
// ---------------------------------------------------------------------------
__global__ __launch_bounds__(256) void prep_kernel(const float* __restrict__ emb,
                                                   float* __restrict__ ee,
                                                   int* __restrict__ used) {
    int v = blockIdx.x * 256 + threadIdx.x;   // grid covers VOCAB
    const float* e = emb + (size_t)v * DIM;
    float s = 0.0f;
#pragma unroll
    for (int c = 0; c < DIM; ++c) s += e[c] * e[c];
    ee[v]   = s;
    used[v] = 0;
}

// ---------------------------------------------------------------------------
// Kernel 2: WMMA fp32 nearest-codebook argmin with double-buffered async
//           global->LDS staging of the codebook chunks.
//   score(row, col) = ||e_col||^2 - 2 * (q_row . e_col)    (||q||^2 dropped)
//   One wave = one 16-row tile; 512 chunks of 16 codes; chunk i+1 streams into
//   LDS (ASYNCcnt) while chunk i feeds the v_wmma_f32_16x16x4_f32 chain.
// ---------------------------------------------------------------------------
__global__ __launch_bounds__(256) void argmin_wmma_kernel(
    const float* __restrict__ f,    // [B,C,T,H,W]
    const float* __restrict__ emb,  // [V, C]
    const float* __restrict__ ee,   // [V]
    int* __restrict__ idx,          // [NQ]
    int* __restrict__ used)         // [V]
{
    __shared__ float smem[2][16 * DIM];   // double-buffered chunk: 2 x 2KB

    const int tid  = threadIdx.x;
    const int lane = tid & 31;
    const int wave = tid >> 5;
    const int tile = blockIdx.x * 8 + wave;   // which 16-row M tile
    const int l15  = lane & 15;
    const int hi   = lane >> 4;               // 0 or 1 (lane half)

    // ---- load A fragments (16 rows x 32 K, fp32) once ----
    // q[n][c] = f[b, c, t, h, w] with n = b*THW + thw; f index = (b*32+c)*THW + thw
    const int r   = tile * 16 + l15;          // this lane's query row (A: M = lane&15)
    const int b   = r >> 13;                  // r / THW
    const int thw = r & (THW - 1);
    const float* fb = f + (size_t)b * DIM * THW + thw;

    v2f a[8];
#pragma unroll
    for (int s = 0; s < 8; ++s) {
        const int k = 4 * s + 2 * hi;         // A: vgpr0 -> K, vgpr1 -> K+1
        a[s].x = fb[(size_t)k * THW];
        a[s].y = fb[(size_t)(k + 1) * THW];
    }

    float best[8];
    int   bi[8];
#pragma unroll
    for (int u = 0; u < 8; ++u) { best[u] = 3.4e38f; bi[u] = 0; }

    // Each thread DMAs 8 bytes of the 2KB chunk (256 threads cover it once).
    const unsigned my8    = (unsigned)tid * 8u;
    const unsigned ldsbuf0 = lds_addr_of(&smem[0][0]) + my8;
    const unsigned ldsbuf1 = lds_addr_of(&smem[1][0]) + my8;

    // Prologue: start chunk 0 into buffer 0.
    async_stage_b64(ldsbuf0, emb, my8);

    for (int i = 0; i < NCHUNK; ++i) {
        const int cur = i & 1;

        wait_async0();        // my piece of chunk i has landed in LDS
        __syncthreads();      // everyone's piece landed; chunk i-1 reads drained

        // Kick chunk i+1 into the other buffer; overlaps with the WMMA chain.
        if (i + 1 < NCHUNK) {
            async_stage_b64(cur ? ldsbuf0 : ldsbuf1, emb,
                            (unsigned)(i + 1) * 2048u + my8);
        }

        const float* buf = &smem[cur][0];
        v8f acc = {0.f, 0.f, 0.f, 0.f, 0.f, 0.f, 0.f, 0.f};
#pragma unroll
        for (int s = 0; s < 8; ++s) {
            const int k = 4 * s + 2 * hi;     // B: N = lane&15, K = 2*hi + vgpr
            v2f bfrag;
            bfrag.x = buf[l15 * DIM + k];
            bfrag.y = buf[l15 * DIM + k + 1];
            // D = A(16x4) x B(4x16) + C  -- exact fp32
            acc = __builtin_amdgcn_wmma_f32_16x16x4_f32(
                false, a[s], false, bfrag, (short)0, acc, false, false);
        }

        const int   col = i * 16 + l15;       // C/D: N = lane&15
        const float en  = ee[col];
#pragma unroll
        for (int u = 0; u < 8; ++u) {         // C/D: M = u + 8*hi
            const float score = en - 2.0f * acc[u];
            if (score < best[u]) { best[u] = score; bi[u] = col; }
        }
    }

    // ---- min-reduce across the 16 lanes of each half (tie -> lower index) ----
#pragma unroll
    for (int u = 0; u < 8; ++u) {
#pragma unroll
        for (int m = 1; m < 16; m <<= 1) {
            const float od = __shfl_xor(best[u], m, 32);
            const int   oi = __shfl_xor(bi[u],   m, 32);
            if (od < best[u] || (od == best[u] && oi < bi[u])) {
                best[u] = od; bi[u] = oi;
            }
        }
    }

    if (l15 == 0) {                            // lane 0 -> rows 0..7, lane 16 -> rows 8..15
        const int rbase = tile * 16 + hi * 8;
#pragma unroll
        for (int u = 0; u < 8; ++u) {
            idx[rbase + u] = bi[u];
            used[bi[u]] = 1;                   // benign race: all writers store 1
        }
    }
}

// ---------------------------------------------------------------------------
// Kernel 3: gather fh = emb[idx], 3x3x3 SAME conv, 50/50 blend, write fhat,
//           per-block deterministic partial sum of squared error.
//   linear = (((b*32+c)*8+t)*32+h)*32+w ; one block = one (b,c,t) x 8h x 32w
// ---------------------------------------------------------------------------
__global__ __launch_bounds__(256) void conv_blend_kernel(
    const float* __restrict__ f,
    const float* __restrict__ emb,
    const float* __restrict__ cw,    // [32,32,3,3,3]
    const float* __restrict__ cb,    // [32]
    const int* __restrict__ idx,     // [NQ]
    float* __restrict__ out,         // [NELEM (+2)]
    float* __restrict__ partial)     // [gridDim.x]
{
    __shared__ float wsm[32 * 27];   // weights for this block's out-channel
    __shared__ float red[256];

    const int tid    = threadIdx.x;
    const int linear = blockIdx.x * 256 + tid;
    const int w = linear & 31;
    const int h = (linear >> 5) & 31;
    const int t = (linear >> 10) & 7;
    const int c = (linear >> 13) & 31;   // uniform within block
    const int b = linear >> 18;

    const float* wc = cw + (size_t)c * 32 * 27;
    for (int i = tid; i < 32 * 27; i += 256) wsm[i] = wc[i];
    __syncthreads();

    const int base_n = b * THW;
    float sum = 0.0f;
#pragma unroll
    for (int kt = 0; kt < 3; ++kt) {
        const int tt = t + kt - 1;
        if (tt < 0 || tt >= 8) continue;
#pragma unroll
        for (int kh = 0; kh < 3; ++kh) {
            const int hh = h + kh - 1;
            if (hh < 0 || hh >= 32) continue;
#pragma unroll
            for (int kw = 0; kw < 3; ++kw) {
                const int ww = w + kw - 1;
                if (ww < 0 || ww >= 32) continue;
                const int code = idx[base_n + tt * 1024 + hh * 32 + ww];
                const float* ev = emb + (size_t)code * DIM;
                const float* wv = wsm + kt * 9 + kh * 3 + kw;  // stride 27 over ci
#pragma unroll
                for (int ci = 0; ci < DIM; ++ci)
                    sum += ev[ci] * wv[ci * 27];
            }
        }
    }

    const int   code_c = idx[base_n + t * 1024 + h * 32 + w];
    const float fh_c   = emb[(size_t)code_c * DIM + c];
    const float val    = 0.5f * fh_c + 0.5f * (sum + cb[c]);   // RESI_RATIO = 0.5
    out[linear] = val;                                          // fhat == fh numerically

    const float d = val - f[linear];
    red[tid] = d * d;
    __syncthreads();
    for (int s = 128; s > 0; s >>= 1) {
        if (tid < s) red[tid] += red[tid + s];
        __syncthreads();
    }
    if (tid == 0) partial[blockIdx.x] = red[0];
}

// ---------------------------------------------------------------------------
// Kernel 4: finalize scalars — vq_loss = 1.25 * SSE/NELEM ;
//           vocab_usage = 100 * (#codes with count>=1)/V   (threshold 0.01/V
//           is below the minimum nonzero prob 1/16384, so count>=1 <=> used)
// ---------------------------------------------------------------------------
__global__ __launch_bounds__(256) void finalize_kernel(
    const float* __restrict__ partial,  // [2048]
    const int* __restrict__ used,       // [V]
    float* __restrict__ out)
{
    __shared__ float rf[256];
    __shared__ int   ri[256];
    const int tid = threadIdx.x;
    float s = 0.0f;
    for (int i = tid; i < 2048; i += 256) s += partial[i];
    int u = 0;
    for (int i = tid; i < VOCAB; i += 256) u += used[i];
    rf[tid] = s;
    ri[tid] = u;
    __syncthreads();
    for (int sh = 128; sh > 0; sh >>= 1) {
        if (tid < sh) { rf[tid] += rf[tid + sh]; ri[tid] += ri[tid + sh]; }
        __syncthreads();
    }
    if (tid == 0) {
        out[NELEM]     = 1.25f * rf[0] / (float)NELEM;          // vq_loss
        out[NELEM + 1] = 100.0f * (float)ri[0] / (float)VOCAB;  // vocab_usage
    }
}

// ---------------------------------------------------------------------------
extern "C" void kernel_launch(void* const* d_in, const int* in_sizes, int n_in,
                              void* d_out, int out_size, void* d_ws, size_t ws_size,
                              hipStream_t stream) {
    const float* f   = (const float*)d_in[0];   // [2,32,8,32,32]
    const float* emb = (const float*)d_in[1];   // [8192,32]
    const float* cw  = (const float*)d_in[2];   // [32,32,3,3,3]
    const float* cb  = (const float*)d_in[3];   // [32]
    float* out = (float*)d_out;                 // [524288 + 2]

    char* ws = (char*)d_ws;
    float* ee      = (float*)ws;                                   //  8192 f
    int*   idx     = (int*)(ws + (size_t)VOCAB * 4);               // 16384 i
    int*   used    = (int*)(ws + (size_t)(VOCAB + NQ) * 4);        //  8192 i
    float* partial = (float*)(ws + (size_t)(2 * VOCAB + NQ) * 4);  //  2048 f

    prep_kernel<<<VOCAB / 256, 256, 0, stream>>>(emb, ee, used);
    argmin_wmma_kernel<<<(NQ / 16) / 8, 256, 0, stream>>>(f, emb, ee, idx, used);
    conv_blend_kernel<<<NELEM / 256, 256, 0, stream>>>(f, emb, cw, cb, idx, out, partial);
    finalize_kernel<<<1, 256, 0, stream>>>(partial, used, out);
}